// VectorQuantizer_38886633898722
// MI455X (gfx1250) — compile-verified
//
#include <hip/hip_runtime.h>
#include <hip/hip_bf16.h>

// ---------------------------------------------------------------------------
// VectorQuantizer forward for MI455X (gfx1250, wave32, WMMA f32 path)
//   B=2 C=16 D=16 H=64 W=64, N_E=2048, E_DIM=16, BETA=0.25
//   T = 131072 tokens, DHW = 65536
//
// Design:
//  - full codebook resident in LDS with row stride 18 dwords (bank-conflict-
//    free for lane-indexed row gathers; slot [16] holds ||e_n||^2)
//  - each wave owns 2 x 16-token tiles -> every codebook B-fragment feeds
//    8 V_WMMA_F32_16X16X4_F32 ops (distance GEMM, f32-exact vs reference)
//  - argmin via per-lane running min + shfl_xor reduction (wave32 halves)
//  - loss from A-fragment registers, one atomicAdd per wave
//  - post_quant_conv via WMMA on LDS-gathered codebook rows
// ---------------------------------------------------------------------------

typedef __attribute__((ext_vector_type(2))) float v2f;
typedef __attribute__((ext_vector_type(8))) float v8f;

#define VQ_T     131072
#define VQ_DHW   65536
#define VQ_NE    2048
#define VQ_E     16
#define EMBS     18        // padded LDS row stride in dwords
#define VQ_WAVES 8
#define MT       2         // 16-token tiles per wave

__global__ void vq_zero_loss(float* loss) { *loss = 0.0f; }

__global__ __launch_bounds__(256) void vq_main_kernel(
    const float* __restrict__ z,        // [2,16,16,64,64]
    const float* __restrict__ qw,       // [16,16]  quant_w[e][c]
    const float* __restrict__ qb,       // [16]
    const float* __restrict__ emb,      // [2048,16]
    const float* __restrict__ pw,       // [16,16]  post_w[c][e]
    const float* __restrict__ pb,       // [16]
    float* __restrict__ out,            // [2,16,16,64,64]
    float* __restrict__ loss,           // [1]
    int*   __restrict__ idx_out)        // [131072]
{
    __shared__ __align__(16) float lds_emb[VQ_NE * EMBS];        // 144 KB
    __shared__ __align__(16) float lds_tr[VQ_WAVES][16 * EMBS];  //   9 KB
    __shared__ int lds_idx[VQ_WAVES][MT * 16];                   //   1 KB

    const int tid = threadIdx.x;

    // ---- stage codebook: row n from 64B-coalesced global, padded LDS rows
#pragma unroll
    for (int i = 0; i < 8; ++i) {
        const int n = tid + i * 256;
        const float4* g = (const float4*)(emb + n * VQ_E);
        const float4 r0 = g[0], r1 = g[1], r2 = g[2], r3 = g[3];
        float* d = &lds_emb[n * EMBS];
        d[0] = r0.x;  d[1] = r0.y;  d[2]  = r0.z;  d[3]  = r0.w;
        d[4] = r1.x;  d[5] = r1.y;  d[6]  = r1.z;  d[7]  = r1.w;
        d[8] = r2.x;  d[9] = r2.y;  d[10] = r2.z;  d[11] = r2.w;
        d[12] = r3.x; d[13] = r3.y; d[14] = r3.z;  d[15] = r3.w;
    }
    __syncthreads();
    // ---- ||e_n||^2 into padding slot [16]
#pragma unroll
    for (int i = 0; i < 8; ++i) {
        const int n = tid + i * 256;
        const float* r = &lds_emb[n * EMBS];
        float s = 0.0f;
#pragma unroll
        for (int e = 0; e < VQ_E; ++e) s += r[e] * r[e];
        lds_emb[n * EMBS + 16] = s;
    }
    __syncthreads();

    const int wave = tid >> 5;
    const int lane = tid & 31;
    const int l    = lane & 15;
    const int half = lane >> 4;
    const int h2   = 2 * half;

    const int wt  = blockIdx.x * VQ_WAVES + wave;   // 4096 wave-tiles x 32 tokens
    const int t0  = wt * 32;
    const int bb  = t0 >> 16;                       // t0 / DHW
    const int sp0 = t0 & 0xFFFF;

    float* trs  = &lds_tr[wave][0];
    int*   idxs = &lds_idx[wave][0];

    // =====================================================================
    // 1) quant_conv for both tiles; keep zf as distance-GEMM A-fragments
    // =====================================================================
    v2f ad[MT][4];
    const float qbias = qb[l];
#pragma unroll
    for (int tt = 0; tt < MT; ++tt) {
        v8f acc;
#pragma unroll
        for (int r = 0; r < 8; ++r) acc[r] = qbias;
        const float* zb = z + (size_t)bb * (16 * VQ_DHW) + sp0 + tt * 16 + l;
#pragma unroll
        for (int kb = 0; kb < 4; ++kb) {
            const int c0 = kb * 4 + h2;
            v2f a, b;
            a.x = zb[(size_t)c0 * VQ_DHW];
            a.y = zb[(size_t)(c0 + 1) * VQ_DHW];
            b.x = qw[l * 16 + c0];
            b.y = qw[l * 16 + c0 + 1];
            acc = __builtin_amdgcn_wmma_f32_16x16x4_f32(false, a, false, b,
                                                        (short)0, acc, false, false);
        }
        // C/D layout (vgpr r -> M=r+8*half, N=l) -> A layout via padded LDS
#pragma unroll
        for (int r = 0; r < 8; ++r) trs[(r + 8 * half) * EMBS + l] = acc[r];
#pragma unroll
        for (int kb = 0; kb < 4; ++kb) {
            const int e0 = kb * 4 + h2;
            ad[tt][kb].x = trs[l * EMBS + e0];
            ad[tt][kb].y = trs[l * EMBS + e0 + 1];
        }
    }

    // =====================================================================
    // 2) distances + argmin: score = ||e_n||^2 - 2 * zf.e_n
    //    each B-fragment feeds 8 WMMAs (2 token tiles)
    // =====================================================================
    float minv[MT][8];
    int   mini[MT][8];
#pragma unroll
    for (int tt = 0; tt < MT; ++tt)
#pragma unroll
        for (int r = 0; r < 8; ++r) { minv[tt][r] = 3.4e38f; mini[tt][r] = 0; }

    for (int nt = 0; nt < VQ_NE / 16; ++nt) {
        const int nb = nt * 16;
        const float* er = &lds_emb[(nb + l) * EMBS];   // conflict-free: stride 18
        v2f bd[4];
#pragma unroll
        for (int kb = 0; kb < 4; ++kb) {
            const int e0 = kb * 4 + h2;
            bd[kb].x = er[e0];
            bd[kb].y = er[e0 + 1];
        }
        const float en = er[16];
        const int   n  = nb + l;
        v8f s0 = {}, s1 = {};
#pragma unroll
        for (int kb = 0; kb < 4; ++kb)
            s0 = __builtin_amdgcn_wmma_f32_16x16x4_f32(false, ad[0][kb], false, bd[kb],
                                                       (short)0, s0, false, false);
#pragma unroll
        for (int kb = 0; kb < 4; ++kb)
            s1 = __builtin_amdgcn_wmma_f32_16x16x4_f32(false, ad[1][kb], false, bd[kb],
                                                       (short)0, s1, false, false);
#pragma unroll
        for (int r = 0; r < 8; ++r) {
            const float d0 = en - 2.0f * s0[r];
            if (d0 < minv[0][r]) { minv[0][r] = d0; mini[0][r] = n; }
            const float d1 = en - 2.0f * s1[r];
            if (d1 < minv[1][r]) { minv[1][r] = d1; mini[1][r] = n; }
        }
    }

    // ---- cross-lane argmin (masks 1..8 stay inside each 16-lane half);
    //      tie-break to lowest index to match jnp.argmin
#pragma unroll
    for (int tt = 0; tt < MT; ++tt)
#pragma unroll
        for (int r = 0; r < 8; ++r) {
#pragma unroll
            for (int m = 1; m <= 8; m <<= 1) {
                const float ov = __shfl_xor(minv[tt][r], m, 32);
                const int   oi = __shfl_xor(mini[tt][r], m, 32);
                if (ov < minv[tt][r] || (ov == minv[tt][r] && oi < mini[tt][r])) {
                    minv[tt][r] = ov; mini[tt][r] = oi;
                }
            }
        }
    if (l == 0) {
#pragma unroll
        for (int tt = 0; tt < MT; ++tt)
#pragma unroll
            for (int r = 0; r < 8; ++r) {
                const int M = r + 8 * half;
                idxs[tt * 16 + M]            = mini[tt][r];
                idx_out[t0 + tt * 16 + M]    = mini[tt][r];
            }
    }

    // =====================================================================
    // 3) loss = 1.25 * mean((emb[idx]-zf)^2), zf taken from A-fragment regs
    // =====================================================================
    float part = 0.0f;
#pragma unroll
    for (int tt = 0; tt < MT; ++tt) {
        const int ix = idxs[tt * 16 + l];
        const float* er = &lds_emb[ix * EMBS];
#pragma unroll
        for (int kb = 0; kb < 4; ++kb) {
            const int e0 = kb * 4 + h2;
            const float d0 = ad[tt][kb].x - er[e0];
            const float d1 = ad[tt][kb].y - er[e0 + 1];
            part += d0 * d0 + d1 * d1;
        }
    }
#pragma unroll
    for (int m = 1; m <= 16; m <<= 1) part += __shfl_xor(part, m, 32);
    if (lane == 0)
        atomicAdd(loss, part * (1.25f / ((float)VQ_T * (float)VQ_E)));

    // =====================================================================
    // 4) post_quant_conv: out[token, cout] = emb[idx] . pw[cout] + pb
    // =====================================================================
    const float pbias = pb[l];
    v2f bo[4];
#pragma unroll
    for (int kb = 0; kb < 4; ++kb) {
        const int e0 = kb * 4 + h2;
        bo[kb].x = pw[l * 16 + e0];
        bo[kb].y = pw[l * 16 + e0 + 1];
    }
#pragma unroll
    for (int tt = 0; tt < MT; ++tt) {
        const int ix = idxs[tt * 16 + l];
        const float* er = &lds_emb[ix * EMBS];
        v8f ao;
#pragma unroll
        for (int r = 0; r < 8; ++r) ao[r] = pbias;
#pragma unroll
        for (int kb = 0; kb < 4; ++kb) {
            const int e0 = kb * 4 + h2;
            v2f a;
            a.x = er[e0];
            a.y = er[e0 + 1];
            ao = __builtin_amdgcn_wmma_f32_16x16x4_f32(false, a, false, bo[kb],
                                                       (short)0, ao, false, false);
        }
        // transpose tile cout-major through padded scratch, then 32B stores
#pragma unroll
        for (int r = 0; r < 8; ++r) trs[l * EMBS + (r + 8 * half)] = ao[r];
        {
            const int cout = lane >> 1;
            const int so   = (lane & 1) * 8;
            const float* srcp = &trs[cout * EMBS + so];
            float tmp[8];
#pragma unroll
            for (int k = 0; k < 8; ++k) tmp[k] = srcp[k];
            float4 p0, p1;
            p0.x = tmp[0]; p0.y = tmp[1]; p0.z = tmp[2]; p0.w = tmp[3];
            p1.x = tmp[4]; p1.y = tmp[5]; p1.z = tmp[6]; p1.w = tmp[7];
            float* dst = out + (size_t)bb * (16 * VQ_DHW) + (size_t)cout * VQ_DHW
                             + sp0 + tt * 16 + so;
            *(float4*)(dst)     = p0;
            *(float4*)(dst + 4) = p1;
        }
    }
}

extern "C" void kernel_launch(void* const* d_in, const int* in_sizes, int n_in,
                              void* d_out, int out_size, void* d_ws, size_t ws_size,
                              hipStream_t stream) {
    const float* z   = (const float*)d_in[0];
    const float* qw  = (const float*)d_in[1];
    const float* qb  = (const float*)d_in[2];
    const float* emb = (const float*)d_in[3];
    const float* pw  = (const float*)d_in[4];
    const float* pb  = (const float*)d_in[5];

    float* outp   = (float*)d_out;            // [2097152] conv output
    float* loss_p = outp + 2 * 16 * VQ_DHW;   // [1] loss
    int*   idx_p  = (int*)(loss_p + 1);       // [131072] int32 indices

    vq_zero_loss<<<1, 1, 0, stream>>>(loss_p);

    const int wave_tiles = VQ_T / (MT * 16);        // 4096
    const int blocks     = wave_tiles / VQ_WAVES;   // 512
    vq_main_kernel<<<blocks, 256, 0, stream>>>(z, qw, qb, emb, pw, pb,
                                               outp, loss_p, idx_p);
}